// MultiVariatePoly_79431125172612
// MI455X (gfx1250) — compile-verified
//
#include <hip/hip_runtime.h>
#include <hip/hip_bf16.h>

// CDNA5 / gfx1250: out[i] = <kron(P(x_i,0..3)), W> + b via the bilinear
// factorization out_i = u_i^T * Wmat * v_i, batched 16 samples per wave32
// into a 64x64x16 fp32 GEMM on V_WMMA_F32_16X16X4_F32.
//
// Layout (CDNA5 ISA 7.12.2, 32-bit operands, wave32):
//   A 16x4 : v0 lane L -> (M=L&15, K=2*(L>>4)),  v1 -> K+1
//   B 4x16 : v0 lane L -> (K=2*(L>>4), N=L&15),  v1 -> K+1
//   C/D 16x16: vgpr r lane L -> (M=r+8*(L>>4), N=L&15)

typedef __attribute__((ext_vector_type(2))) float v2f;
typedef __attribute__((ext_vector_type(8))) float v8f;

#define POLY_ORDER 7
#define NP 8          // polynomials per dim (order+1)
#define MT 4          // M tiles: 64 rows / 16
#define KC 16         // K chunks: 64 cols / 4

__global__ __launch_bounds__(256) void poly_wmma_kernel(
    const float* __restrict__ x,      // N x 4
    const float* __restrict__ W,      // 4096 = 64 (u) x 64 (v), row-major
    const float* __restrict__ bias_p, // 1
    float* __restrict__ out,          // N
    int N, int groups_per_wave)
{
  const int lane = threadIdx.x & 31;
  const int wave = threadIdx.x >> 5;
  const int wave_global = blockIdx.x * (blockDim.x >> 5) + wave;
  const bool hi = (lane >= 16);   // lane half
  const int sl = lane & 15;       // sample slot within 16-sample group
  const int h2 = hi ? 2 : 0;      // 2*h, used only for A addressing

  // ---- Load A = Wmat tiles into registers, once per wave (W is 16KB, L2-hot).
  // A[m][k].x = Wmat[16m + sl][4k + 2h], .y = next K column.
  v2f A[MT][KC];
#pragma unroll
  for (int m = 0; m < MT; ++m) {
    const float* wrow = W + (16 * m + sl) * 64 + h2;
#pragma unroll
    for (int k = 0; k < KC; ++k) {
      A[m][k] = *(const v2f*)(wrow + 4 * k);
    }
  }
  const float bias = bias_p[0];

  for (int g = 0; g < groups_per_wave; ++g) {
    const int s0 = (wave_global * groups_per_wave + g) * 16;  // wave-uniform
    if (s0 >= N) break;                                       // uniform: EXEC stays full
    const int s = s0 + sl;

    // ---- Legendre P0..P7, all 4 dims, Bonnet recurrence.
    // Division replaced by compile-time reciprocal constants (no v_div_* path).
    const float4 xs = *(const float4*)(x + s * 4);
    float xv[4] = {xs.x, xs.y, xs.z, xs.w};
    float leg[4][NP];
#pragma unroll
    for (int d = 0; d < 4; ++d) {
      leg[d][0] = 1.0f;
      leg[d][1] = xv[d];
#pragma unroll
      for (int n = 1; n < POLY_ORDER; ++n) {
        const float rn = 1.0f / (float)(n + 1);   // compile-time constant
        leg[d][n + 1] =
            ((2.0f * n + 1.0f) * xv[d] * leg[d][n] - (float)n * leg[d][n - 1]) * rn;
      }
    }

    // ---- Pre-select the lane-half-dependent values ONCE (one cndmask each),
    // so the k-loop uses only compile-time register indices.
    // B chunk k needs leg3[(k&1)*4 + 2h + t], t=0,1  ->  4 values per lane:
    float l3[4];
    l3[0] = hi ? leg[3][2] : leg[3][0];
    l3[1] = hi ? leg[3][3] : leg[3][1];
    l3[2] = hi ? leg[3][6] : leg[3][4];
    l3[3] = hi ? leg[3][7] : leg[3][5];
    // Final reduction needs leg0[2m+h] -> 4 values per lane:
    float l0[MT];
#pragma unroll
    for (int m = 0; m < MT; ++m) l0[m] = hi ? leg[0][2 * m + 1] : leg[0][2 * m];

    // ---- Z(64x16) = Wmat(64x64) @ V(64x16), K swept in 16 chunks of 4.
    v8f acc[MT];
#pragma unroll
    for (int m = 0; m < MT; ++m)
      acc[m] = (v8f){0.f, 0.f, 0.f, 0.f, 0.f, 0.f, 0.f, 0.f};

#pragma unroll
    for (int k = 0; k < KC; ++k) {
      // v[idx] = P_c(x2)*P_d(x3); within chunk k, c = k>>1 is constant.
      const float c2 = leg[2][k >> 1];
      const int o = (k & 1) * 2;          // selects {l3[0],l3[1]} or {l3[2],l3[3]}
      v2f B;
      B.x = c2 * l3[o];
      B.y = c2 * l3[o + 1];
#pragma unroll
      for (int m = 0; m < MT; ++m) {
        acc[m] = __builtin_amdgcn_wmma_f32_16x16x4_f32(
            /*neg_a=*/false, A[m][k], /*neg_b=*/false, B,
            /*c_mod=*/(short)0, acc[m], /*reuse_a=*/false, /*reuse_b=*/false);
      }
    }

    // ---- out_i = u_i . Z[:,i]: lane holds Z rows {16m+8h+r};
    // row 16m+8h+r -> u index a=2m+h, b=r -> l0[m]*leg1[r].
    float partial = 0.0f;
#pragma unroll
    for (int m = 0; m < MT; ++m) {
      float t = 0.0f;
#pragma unroll
      for (int r = 0; r < 8; ++r) t = fmaf(leg[1][r], acc[m][r], t);
      partial = fmaf(l0[m], t, partial);
    }
    // merge the two lane-halves covering the same sample
    partial += __shfl_xor(partial, 16, 32);
    if (!hi) out[s] = partial + bias;
  }
}

extern "C" void kernel_launch(void* const* d_in, const int* in_sizes, int n_in,
                              void* d_out, int out_size, void* d_ws, size_t ws_size,
                              hipStream_t stream) {
  (void)n_in; (void)d_ws; (void)ws_size; (void)out_size;
  const float* x = (const float*)d_in[0];   // N x 4 f32
  const float* W = (const float*)d_in[1];   // 4096 f32
  const float* b = (const float*)d_in[2];   // 1 f32
  float* out = (float*)d_out;               // N f32

  const int N = in_sizes[0] / 4;
  const int WAVES_PER_BLOCK = 8;            // 256 threads, wave32
  const int GROUPS_PER_WAVE = 2;            // 2 * 16 = 32 samples per wave
  const int samples_per_block = WAVES_PER_BLOCK * 16 * GROUPS_PER_WAVE;
  const int blocks = (N + samples_per_block - 1) / samples_per_block;

  poly_wmma_kernel<<<blocks, WAVES_PER_BLOCK * 32, 0, stream>>>(
      x, W, b, out, N, GROUPS_PER_WAVE);
}